// FilterDetections_29566554866143
// MI455X (gfx1250) — compile-verified
//
#include <hip/hip_runtime.h>
#include <cstdint>
#include <cstddef>

#define B_ 16
#define N_ 50000
#define C_ 20
#define MAXDET 100
#define SCORE_THR_F 0.01f
#define IOU_THR_F 0.5f
#define NEGV -1e30f
#define K1_THREADS 1024
#define K2_THREADS 256
#define USE_TDM 1

typedef __attribute__((ext_vector_type(4))) unsigned int u32x4_t;
typedef __attribute__((ext_vector_type(8))) unsigned int u32x8_t;

// ---- packed (score, idx) key: larger key == higher score, tie -> lower idx ----
__device__ __forceinline__ unsigned long long packKey(float s, int idx) {
    unsigned u = __float_as_uint(s);
    u = (u & 0x80000000u) ? ~u : (u | 0x80000000u);   // order-preserving map
    return ((unsigned long long)u << 32) | (unsigned)(~idx);
}
__device__ __forceinline__ int keyIdx(unsigned long long k) {
    return (int)(~(unsigned)(k & 0xFFFFFFFFull));
}
__device__ __forceinline__ float keyScore(unsigned long long k) {
    unsigned u = (unsigned)(k >> 32);
    u = (u & 0x80000000u) ? (u ^ 0x80000000u) : ~u;
    return __uint_as_float(u);
}

// ---- wave32 + cross-wave max reduction ----
__device__ __forceinline__ unsigned long long waveMax(unsigned long long k) {
    for (int off = 16; off > 0; off >>= 1) {
        unsigned long long o = __shfl_xor(k, off, 32);
        if (o > k) k = o;
    }
    return k;
}

__device__ __forceinline__ unsigned long long blockMax(unsigned long long k,
                                                       unsigned long long* s_red,
                                                       unsigned long long* s_b,
                                                       int nWaves) {
    const int lane = threadIdx.x & 31;
    const int wid  = threadIdx.x >> 5;
    k = waveMax(k);
    __syncthreads();                 // protect s_red reuse across iterations
    if (lane == 0) s_red[wid] = k;
    __syncthreads();
    if (wid == 0) {
        unsigned long long kk = (lane < nWaves) ? s_red[lane] : 0ull;
        kk = waveMax(kk);
        if (lane == 0) *s_b = kk;
    }
    __syncthreads();
    return *s_b;
}

// ---- IOU exactly mirroring the reference (no FMA contraction) ----
__device__ __forceinline__ float iou_ref(float4 a, float4 b) {
#pragma clang fp contract(off)
    float ix1 = fmaxf(a.x, b.x), iy1 = fmaxf(a.y, b.y);
    float ix2 = fminf(a.z, b.z), iy2 = fminf(a.w, b.w);
    float inter = fmaxf(ix2 - ix1, 0.0f) * fmaxf(iy2 - iy1, 0.0f);
    float aa = fmaxf(a.z - a.x, 0.0f) * fmaxf(a.w - a.y, 0.0f);
    float ab = fmaxf(b.z - b.x, 0.0f) * fmaxf(b.w - b.y, 0.0f);
    float un = aa + ab - inter;
    return (un > 0.0f) ? (inter / un) : 0.0f;
}

// =====================================================================
// Kernel 1: one workgroup per (b, c). Scores live in LDS (200 KB).
// TDM gathers the strided class column into LDS; then 100 fused
// argmax+suppress sweeps. Boxes stream from L2 (12.8 MB resident).
// =====================================================================
__global__ __launch_bounds__(K1_THREADS) void nms_k(
        const float4* __restrict__ boxes4,
        const float*  __restrict__ cls,
        float* __restrict__ cand_score,
        int*   __restrict__ cand_idx) {
    __shared__ float s_sc[N_];                         // 200000 B
    __shared__ unsigned long long s_red[K1_THREADS / 32];
    __shared__ unsigned long long s_best;

    const int b   = blockIdx.x / C_;
    const int c   = blockIdx.x % C_;
    const int tid = threadIdx.x;
    const size_t bbase = (size_t)b * N_;
    const float* col = cls + ((size_t)b * N_) * C_ + c;

#if USE_TDM
    // --- Tensor Data Mover: 2D descriptor, tile 1 x 50000, row stride C elems ---
    if (tid < 32) {  // wave 0 only (TDM is a per-wave DMA op; EXEC ignored)
        const uint64_t ga  = (uint64_t)(uintptr_t)col;
        const uint32_t lds = (uint32_t)(uintptr_t)(&s_sc[0]);
        u32x4_t g0;
        g0[0] = 1u;                                         // count=1, user mode
        g0[1] = lds;                                        // lds_addr
        g0[2] = (uint32_t)ga;                               // global_addr[31:0]
        g0[3] = ((uint32_t)(ga >> 32) & 0x01FFFFFFu)        // global_addr[56:32]
              | 0x80000000u;                                // type=2 ("image")
        u32x8_t g1;
        g1[0] = 0x00020000u;                                // data_size=2 (4B)
        g1[1] = 0x00010000u;                                // tensor_dim0 = 1
        g1[2] = ((uint32_t)N_ & 0xFFFFu) << 16;             // tensor_dim1 lo16
        g1[3] = 0x00010000u;                                // tile_dim0 = 1
        g1[4] = (uint32_t)N_ & 0xFFFFu;                     // tile_dim1 = 50000
        g1[5] = (uint32_t)C_;                               // dim0_stride = 20
        g1[6] = 0u;
        g1[7] = 0u;
        asm volatile("tensor_load_to_lds %0, %1" :: "s"(g0), "s"(g1) : "memory");
        __builtin_amdgcn_s_wait_tensorcnt(0);
    }
    __syncthreads();
#else
    for (int i = tid; i < N_; i += K1_THREADS) s_sc[i] = col[(size_t)i * C_];
    __syncthreads();
#endif

    // --- threshold + initial argmax in one pass ---
    unsigned long long local = 0;
    for (int i = tid; i < N_; i += K1_THREADS) {
        float v = s_sc[i];
        if (v > SCORE_THR_F) {
            unsigned long long k = packKey(v, i);
            if (k > local) local = k;
        } else {
            s_sc[i] = NEGV;
        }
    }
    unsigned long long bestKey = blockMax(local, s_red, &s_best, K1_THREADS / 32);

    const unsigned thrBits = __float_as_uint(SCORE_THR_F) | 0x80000000u;
    const size_t cbase = (size_t)blockIdx.x * MAXDET;

    for (int d = 0; d < MAXDET; ++d) {
        const bool ok = (unsigned)(bestKey >> 32) > thrBits;  // strict s > thr
        if (!ok) {
            for (int dd = d + tid; dd < MAXDET; dd += K1_THREADS) {
                cand_score[cbase + dd] = NEGV;
                cand_idx[cbase + dd]   = 0;
            }
            break;
        }
        const int j = keyIdx(bestKey);
        if (tid == 0) {
            cand_score[cbase + d] = keyScore(bestKey);
            cand_idx[cbase + d]   = j;
        }
        if (d == MAXDET - 1) break;

        // --- fused suppression + next argmax (single sweep) ---
        const float4 bj = boxes4[bbase + j];                 // broadcast L2 hit
        unsigned long long loc = 0;
        for (int i = tid; i < N_; i += K1_THREADS) {
            float s = s_sc[i];
            if (s > -1e29f) {                                // alive only
                float4 bi = boxes4[bbase + i];
                if (i == j || iou_ref(bj, bi) > IOU_THR_F) {
                    s_sc[i] = NEGV;
                } else {
                    unsigned long long k = packKey(s, i);
                    if (k > loc) loc = k;
                }
            }
        }
        bestKey = blockMax(loc, s_red, &s_best, K1_THREADS / 32);
    }
}

// =====================================================================
// Kernel 2: one workgroup per batch; stable top-100 of the 2000
// candidates (key scheme reproduces lax.top_k tie-breaking), then emit
// boxes / scores / labels with the valid-mask semantics.
// =====================================================================
__global__ __launch_bounds__(K2_THREADS) void topk_k(
        const float4* __restrict__ boxes4,
        const float*  __restrict__ cand_score,
        const int*    __restrict__ cand_idx,
        float* __restrict__ out) {
    constexpr int P = C_ * MAXDET;            // 2000
    __shared__ unsigned long long keys[P];    // 16 KB
    __shared__ unsigned long long s_red[K2_THREADS / 32];
    __shared__ unsigned long long s_best;

    const int b   = blockIdx.x;
    const int tid = threadIdx.x;

    for (int p = tid; p < P; p += K2_THREADS)
        keys[p] = packKey(cand_score[(size_t)b * P + p], p);
    __syncthreads();

    float* boxes_out  = out;                       // [B,100,4]
    float* scores_out = out + (size_t)B_ * MAXDET * 4;
    float* labels_out = out + (size_t)B_ * MAXDET * 5;

    for (int d = 0; d < MAXDET; ++d) {
        unsigned long long loc = 0;
        for (int p = tid; p < P; p += K2_THREADS) {
            unsigned long long k = keys[p];
            if (k > loc) loc = k;
        }
        unsigned long long best = blockMax(loc, s_red, &s_best, K2_THREADS / 32);
        if (tid == 0) {
            const int   pos   = keyIdx(best);
            const float sc    = keyScore(best);
            const bool  valid = sc > (NEGV * 0.5f);
            const int   anch  = cand_idx[(size_t)b * P + pos];
            const int   o     = b * MAXDET + d;
            float4 bx = make_float4(-1.0f, -1.0f, -1.0f, -1.0f);
            if (valid) bx = boxes4[(size_t)b * N_ + anch];
            boxes_out[(size_t)o * 4 + 0] = bx.x;
            boxes_out[(size_t)o * 4 + 1] = bx.y;
            boxes_out[(size_t)o * 4 + 2] = bx.z;
            boxes_out[(size_t)o * 4 + 3] = bx.w;
            scores_out[o] = valid ? sc : -1.0f;
            labels_out[o] = valid ? (float)(pos / MAXDET) : -1.0f;
            keys[pos] = 0;                                   // remove
        }
        __syncthreads();
    }
}

extern "C" void kernel_launch(void* const* d_in, const int* in_sizes, int n_in,
                              void* d_out, int out_size, void* d_ws, size_t ws_size,
                              hipStream_t stream) {
    const float4* boxes4 = (const float4*)d_in[0];        // (B,N,4) f32
    const float*  cls    = (const float*)d_in[1];         // (B,N,C) f32

    float* cand_score = (float*)d_ws;                                       // B*C*100 f32
    int*   cand_idx   = (int*)((char*)d_ws + (size_t)B_ * C_ * MAXDET * 4); // B*C*100 i32

    nms_k<<<dim3(B_ * C_), dim3(K1_THREADS), 0, stream>>>(boxes4, cls, cand_score, cand_idx);
    topk_k<<<dim3(B_), dim3(K2_THREADS), 0, stream>>>(boxes4, cand_score, cand_idx, (float*)d_out);
}